// RadialMasking_29446295781980
// MI455X (gfx1250) — compile-verified
//
#include <hip/hip_runtime.h>
#include <stdint.h>

// ---------------- constants from the reference ----------------
#define NB        8          // batch frames
#define NGROUPS   360
#define NBINS     4096       // radial histogram bins for nanquantile
#define DMAX      81.0f      // max possible dist = sqrt(70^2+40^2) ~ 80.62
#define P_NEAR    0.48f
#define P_MID     0.80f
#define P_FAR     0.95f
#define SEL_P     0.2f       // 1 - masked_ratio

typedef __attribute__((ext_vector_type(2))) float v2f;
typedef __attribute__((ext_vector_type(8))) float v8f;

// ---------------- JAX threefry2x32 (exact) ----------------
__device__ __forceinline__ uint32_t rotl32(uint32_t v, int r) {
    return (v << r) | (v >> (32 - r));
}

__device__ __forceinline__ void threefry2x32(uint32_t k0, uint32_t k1,
                                             uint32_t x0, uint32_t x1,
                                             uint32_t& o0, uint32_t& o1) {
    uint32_t ks[3] = {k0, k1, k0 ^ k1 ^ 0x1BD11BDAu};
    x0 += ks[0]; x1 += ks[1];
#pragma unroll
    for (int i = 0; i < 5; ++i) {
        if ((i & 1) == 0) {
            x0 += x1; x1 = rotl32(x1, 13); x1 ^= x0;
            x0 += x1; x1 = rotl32(x1, 15); x1 ^= x0;
            x0 += x1; x1 = rotl32(x1, 26); x1 ^= x0;
            x0 += x1; x1 = rotl32(x1, 6);  x1 ^= x0;
        } else {
            x0 += x1; x1 = rotl32(x1, 17); x1 ^= x0;
            x0 += x1; x1 = rotl32(x1, 29); x1 ^= x0;
            x0 += x1; x1 = rotl32(x1, 16); x1 ^= x0;
            x0 += x1; x1 = rotl32(x1, 24); x1 ^= x0;
        }
        x0 += ks[(i + 1) % 3];
        x1 += ks[(i + 2) % 3] + (uint32_t)(i + 1);
    }
    o0 = x0; o1 = x1;
}

__device__ __forceinline__ float u01(uint32_t bits) {
    // JAX: bitcast((bits>>9)|0x3f800000) - 1.0  in [0,1)
    return __uint_as_float((bits >> 9) | 0x3f800000u) - 1.0f;
}

// key = jax.random.key(42) -> (0,42); kg,kv = split(key)
// counts iota(4): pair0=(0,2), pair1=(1,3); kg=(o0p0,o0p1), kv=(o1p0,o1p1)
// All-literal inputs: compiler constant-folds the two threefry calls.
__device__ __forceinline__ void derive_keys(uint32_t& g0, uint32_t& g1,
                                            uint32_t& v0, uint32_t& v1) {
    uint32_t a0, b0, a1, b1;
    threefry2x32(0u, 42u, 0u, 2u, a0, b0);
    threefry2x32(0u, 42u, 1u, 3u, a1, b1);
    g0 = a0; g1 = a1; v0 = b0; v1 = b1;
}

// ---------------- stage-1 per-voxel: dist, batch, keep1 ----------------
__device__ __forceinline__ void voxel_stage1(int4 c, uint32_t g0, uint32_t g1,
                                             int& b, float& dist, bool& keep1) {
    b = c.x;
    float x = (float)c.w * 0.1f - 70.0f;
    float y = (float)c.z * 0.1f - 40.0f;
    float deg = atan2f(y, x) * 57.29577951308232f;   // degrees
    if (deg < 0.0f) deg += 360.0f;                   // mod 360
    int g = (int)deg;
    g = g < 0 ? 0 : (g > NGROUPS - 1 ? NGROUPS - 1 : g);
    dist = sqrtf(x * x + y * y);
    // uniform(kg, (8,360)) flat index t, threefry pairs (t, t+1440)
    int t = b * NGROUPS + g;
    uint32_t lo = (t < 1440) ? (uint32_t)t : (uint32_t)(t - 1440);
    uint32_t o0, o1;
    threefry2x32(g0, g1, lo, lo + 1440u, o0, o1);
    keep1 = u01((t < 1440) ? o0 : o1) < SEL_P;
}

// ---------------- K0: zero histogram scratch ----------------
__global__ void k_zero(uint32_t* p, int n) {
    int i = blockIdx.x * blockDim.x + threadIdx.x;
    if (i < n) p[i] = 0u;
}

// ---------------- K1: radial histogram of stage-1 survivors ----------------
__global__ void k_hist(const int* __restrict__ coords, uint32_t* __restrict__ hist, int N) {
    int i = blockIdx.x * blockDim.x + threadIdx.x;
    if (i >= N) return;
    uint32_t g0, g1, v0, v1;
    derive_keys(g0, g1, v0, v1);
    int4 c = ((const int4*)coords)[i];
    int b; float dist; bool keep1;
    voxel_stage1(c, g0, g1, b, dist, keep1);
    if (keep1) {
        int bin = (int)(dist * ((float)NBINS / DMAX));
        bin = bin > NBINS - 1 ? NBINS - 1 : (bin < 0 ? 0 : bin);
        atomicAdd((unsigned int*)&hist[b * NBINS + bin], 1u);
    }
}

// ---------------- K2: histogram-CDF inversion -> q1,q2 per frame ----------------
__global__ void k_quant(const uint32_t* __restrict__ hist, float* __restrict__ q) {
    int b = threadIdx.x;
    if (b >= NB) return;
    const uint32_t* h = hist + b * NBINS;
    uint32_t M = 0;
    for (int i = 0; i < NBINS; ++i) M += h[i];
    const float binw = DMAX / (float)NBINS;
    float ranks[2] = {0.33f * (float)((int)M - 1), 0.67f * (float)((int)M - 1)};
#pragma unroll
    for (int s = 0; s < 2; ++s) {
        float target = fmaxf(ranks[s], 0.0f);
        uint32_t cacc = 0;
        float val = DMAX;
        for (int i = 0; i < NBINS; ++i) {
            uint32_t cnt = h[i];
            if (cnt != 0u && (float)(cacc + cnt) > target) {
                float p = (target - (float)cacc + 0.5f) / (float)cnt;
                val = ((float)i + fminf(p, 1.0f)) * binw;
                break;
            }
            cacc += cnt;
        }
        q[s * NB + b] = val;
    }
}

// ---------------- K3: keep mask + WMMA-gated feature copy ----------------
// Wave handles 32 voxels (two 16-row tiles). out = diag(keep) x F done with
// v_wmma_f32_16x16x4_f32 chains (full f32 precision), NT hints on the streams.
__global__ void k_gate(const int* __restrict__ coords, const float* __restrict__ feat,
                       const float* __restrict__ q, float* __restrict__ out_keep,
                       float* __restrict__ out_feat, int N) {
    const int lane = threadIdx.x & 31;
    const int wave = threadIdx.x >> 5;
    const int vbase = (blockIdx.x * (blockDim.x >> 5) + wave) * 32;
    if (vbase >= N) return;   // wave-uniform (N % 32 == 0)

    uint32_t g0, g1, vk0, vk1;
    derive_keys(g0, g1, vk0, vk1);

    // --- per-lane voxel: stage1 + stage2 keep ---
    const int i = vbase + lane;
    int4 c = ((const int4*)coords)[i];
    int b; float dist; bool keep1;
    voxel_stage1(c, g0, g1, b, dist, keep1);

    const uint32_t H = (uint32_t)(N >> 1);              // threefry half for uniform(kv,(B,npb))
    uint32_t lo = ((uint32_t)i < H) ? (uint32_t)i : (uint32_t)i - H;
    uint32_t o0, o1;
    threefry2x32(vk0, vk1, lo, lo + H, o0, o1);
    float u = u01(((uint32_t)i < H) ? o0 : o1);

    float q1 = q[b], q2 = q[NB + b];
    float mp = dist < q1 ? P_NEAR : (dist < q2 ? P_MID : P_FAR);
    bool keep = keep1 && (u >= mp);
    float kf = keep ? 1.0f : 0.0f;
    out_keep[i] = kf;

    // keep value for "my" row in each 16-row tile (A/B/C/D half-lane layouts)
    float kswap = __shfl_xor(kf, 16, 32);
    float k_t0 = (lane < 16) ? kf : kswap;   // keep[row=lane&15] of tile 0
    float k_t1 = (lane < 16) ? kswap : kf;   // keep[row=lane&15] of tile 1
    const int m = lane & 15;
    const int half = (lane < 16) ? 0 : 2;    // K parity of this half-wave (A/B layout)

#pragma unroll
    for (int t = 0; t < 2; ++t) {
        const float kt = t ? k_t1 : k_t0;
        const float* Fb = feat + (size_t)(vbase + 16 * t) * 64;
        float* Ob = out_feat + (size_t)(vbase + 16 * t) * 64;
#pragma unroll
        for (int c0 = 0; c0 < 64; c0 += 16) {
            v8f acc = {};
#pragma unroll
            for (int j = 0; j < 4; ++j) {
                const int r0 = 4 * j + half;       // row held in B vgpr0 by this lane
                const int r1 = 4 * j + half + 1;   // row held in B vgpr1
                v2f Bv, Av;
                Bv.x = __builtin_nontemporal_load(&Fb[r0 * 64 + c0 + m]);
                Bv.y = __builtin_nontemporal_load(&Fb[r1 * 64 + c0 + m]);
                // A = columns 4j..4j+3 of diag(keep): A[M][K] = (M==4j+K)?keep[M]:0
                Av.x = (m == r0) ? kt : 0.0f;
                Av.y = (m == r1) ? kt : 0.0f;
                acc = __builtin_amdgcn_wmma_f32_16x16x4_f32(
                    false, Av, false, Bv, (short)0, acc, false, false);
            }
#pragma unroll
            for (int r = 0; r < 8; ++r) {
                const int M = r + ((lane < 16) ? 0 : 8);   // C/D layout
                __builtin_nontemporal_store(acc[r], &Ob[M * 64 + c0 + m]);
            }
        }
    }
}

// ---------------- launch ----------------
extern "C" void kernel_launch(void* const* d_in, const int* in_sizes, int n_in,
                              void* d_out, int out_size, void* d_ws, size_t ws_size,
                              hipStream_t stream) {
    const int* coords = (const int*)d_in[0];     // [N,4] int32
    const float* feat = (const float*)d_in[1];   // [N,64] f32
    const int N = in_sizes[0] / 4;

    uint32_t* hist = (uint32_t*)d_ws;                                  // 8*4096 u32
    float* q = (float*)((char*)d_ws + (size_t)NB * NBINS * sizeof(uint32_t)); // 16 f32

    float* out_keep = (float*)d_out;       // [N] keep as 0/1 f32
    float* out_feat = out_keep + N;        // [N,64] gated features

    const int histn = NB * NBINS;
    k_zero<<<(histn + 255) / 256, 256, 0, stream>>>(hist, histn);
    k_hist<<<(N + 255) / 256, 256, 0, stream>>>(coords, hist, N);
    k_quant<<<1, 64, 0, stream>>>(hist, q);
    // 256 threads = 8 waves = 256 voxels per block; N=1.6M divides exactly
    k_gate<<<(N + 255) / 256, 256, 0, stream>>>(coords, feat, q, out_keep, out_feat, N);
}